// SpecAugmentTransform_24713241821759
// MI455X (gfx1250) — compile-verified
//
#include <hip/hip_runtime.h>
#include <stdint.h>

// ---------------------------------------------------------------------------
// SpecAugment on MI455X (gfx1250). 393 MB single-use stream @ 23.3 TB/s
// => ~17us floor; memory-bound, WMMA irrelevant.
//   kernel A (defined first so it leads the disassembly): per-(b,f)-row
//             masked multiply. Preferred path: Tensor Data Mover
//             (tensor_load_to_lds) double-buffered into LDS with
//             s_wait_tensorcnt pipelining, consume via ds_load_b128, emit
//             nontemporal b128 stores (stream > 192MB L2). Fallback path:
//             direct NT b128 loads + global_prefetch_b8.
//   kernel B: JAX threefry chain (partitionable mode) -> 4 ints / batch row.
// ---------------------------------------------------------------------------

typedef float        f4    __attribute__((ext_vector_type(4)));
typedef unsigned int u32x4 __attribute__((ext_vector_type(4)));
typedef int          i32x4 __attribute__((ext_vector_type(4)));
typedef int          i32x8 __attribute__((ext_vector_type(8)));

#define B_DIM 128
#define F_DIM 128
#define T_DIM 3000
#define F_MAG 40u
#define T_MAG 100u
#define T_VEC (T_DIM / 4)          // 750 float4 per row
#define NROWS (B_DIM * F_DIM)      // 16384 rows
#define APPLY_BLOCKS 2048
#define APPLY_THREADS 256

#ifndef __has_builtin
#define __has_builtin(x) 0
#endif

// Per-float4 time mask * frequency scale.
__device__ __forceinline__ f4 mask4(f4 v, int t, int tlo, int thi, float fs) {
  f4 m;
  m.x = (t     >= tlo && t     < thi) ? 0.0f : fs;
  m.y = (t + 1 >= tlo && t + 1 < thi) ? 0.0f : fs;
  m.z = (t + 2 >= tlo && t + 2 < thi) ? 0.0f : fs;
  m.w = (t + 3 >= tlo && t + 3 < thi) ? 0.0f : fs;
  return v * m;
}

#if defined(__HIP_DEVICE_COMPILE__) && __has_builtin(__builtin_amdgcn_tensor_load_to_lds)
#define SPECAUG_USE_TDM 1
// Issue one TDM descriptor: copy row (3000 fp32, 12 KB) to the LDS address in lds_addr.
// D# packing per CDNA5 ISA 8.3-8.6.
__device__ __forceinline__ void tdm_issue_row(const float* x, int row,
                                              uint32_t lds_addr) {
  const uint64_t ga = (uint64_t)(uintptr_t)(x + (long)row * T_DIM);

  u32x4 g0;
  g0.x = 1u;                                   // count=1, user mode, no gather
  g0.y = lds_addr;                             // lds_addr [63:32]
  g0.z = (uint32_t)ga;                         // global_addr[31:0]
  g0.w = (uint32_t)(ga >> 32) | (2u << 30);    // global_addr[56:32] | type=2

  i32x8 g1;
  g1[0] = (int)(2u << 16);                     // wg_mask=0, data_size=4B
  g1[1] = (int)((uint32_t)T_DIM << 16);        // tensor_dim0[15:0]=3000
  g1[2] = (int)(1u << 16);                     // tensor_dim0[31:16]=0, tensor_dim1=1
  g1[3] = (int)((uint32_t)T_DIM << 16);        // tile_dim0=3000
  g1[4] = 1;                                   // tile_dim1=1, tile_dim2=0
  g1[5] = T_DIM;                               // tensor_dim0_stride=3000
  g1[6] = 0;
  g1[7] = 0;

  const i32x4 z4 = {0, 0, 0, 0};               // groups 2/3: dims unused
#if __clang_major__ >= 23
  const i32x8 z8 = {0, 0, 0, 0, 0, 0, 0, 0};
  __builtin_amdgcn_tensor_load_to_lds(g0, g1, z4, z4, z8, 0);
#else
  __builtin_amdgcn_tensor_load_to_lds(g0, g1, z4, z4, 0);
#endif
}
#else
#define SPECAUG_USE_TDM 0
#endif

// Kernel A: grid-stride over the 16384 (b,f) rows; 256 threads = 8 waves.
__global__ __launch_bounds__(APPLY_THREADS)
void specaug_apply_kernel(const float* __restrict__ x, f4* __restrict__ out,
                          const int* __restrict__ prm) {
#if SPECAUG_USE_TDM
  // TDM double-buffered pipeline: wave 0 issues row r+stride into the spare
  // buffer, then s_wait_tensorcnt 1 (TDM ops of a wave complete in-order, so
  // <=1 outstanding means the current row has landed), barrier, all 8 waves
  // consume from LDS and emit NT b128 stores.
  __shared__ __align__(16) float buf[2][T_DIM];   // 24 KB of 320 KB
  const int stride = (int)gridDim.x;
  const bool issuer = (threadIdx.x < 32);         // wave 0 only
  int r = (int)blockIdx.x;
  int nb = 0;

  if (issuer && r < NROWS)
    tdm_issue_row(x, r, (uint32_t)(uintptr_t)&buf[0][0]);

  for (; r < NROWS; r += stride) {
    const int nxt = r + stride;
    if (issuer) {
      if (nxt < NROWS) {
        tdm_issue_row(x, nxt, (uint32_t)(uintptr_t)&buf[nb ^ 1][0]);
        __builtin_amdgcn_s_wait_tensorcnt(1);     // current row complete
      } else {
        __builtin_amdgcn_s_wait_tensorcnt(0);
      }
    }
    __syncthreads();                              // LDS visible to all waves

    const int b = r >> 7, f = r & (F_DIM - 1);
    const int flo = prm[4 * b + 0], fhi = prm[4 * b + 1];
    const int tlo = prm[4 * b + 2], thi = prm[4 * b + 3];
    const float fs = (f >= flo && f < fhi) ? 0.0f : 1.0f;

    const f4* lb = (const f4*)&buf[nb][0];
    f4* po = out + (long)r * T_VEC;
    for (int i = (int)threadIdx.x; i < T_VEC; i += APPLY_THREADS) {
      f4 v = lb[i];                               // ds_load_b128
      __builtin_nontemporal_store(mask4(v, i << 2, tlo, thi, fs), po + i);
    }
    __syncthreads();                              // buffer free for reuse
    nb ^= 1;
  }
#elif defined(__HIP_DEVICE_COMPILE__)
  // Fallback: direct nontemporal b128 streaming + global_prefetch_b8.
  const int stride = (int)gridDim.x;
  for (int r = (int)blockIdx.x; r < NROWS; r += stride) {
    const int b = r >> 7, f = r & (F_DIM - 1);
    const int flo = prm[4 * b + 0], fhi = prm[4 * b + 1];
    const int tlo = prm[4 * b + 2], thi = prm[4 * b + 3];
    const float fs = (f >= flo && f < fhi) ? 0.0f : 1.0f;

    const f4* px = (const f4*)(x) + (long)r * T_VEC;
    f4* po = out + (long)r * T_VEC;
    for (int i = (int)threadIdx.x; i < T_VEC; i += APPLY_THREADS) {
      __builtin_prefetch((const void*)(px + i + APPLY_THREADS), 0, 3);
      f4 v = __builtin_nontemporal_load(px + i);
      __builtin_nontemporal_store(mask4(v, i << 2, tlo, thi, fs), po + i);
    }
  }
#endif
}

// ----- Threefry-2x32, 20 rounds (JAX's PRNG core) --------------------------
__device__ __forceinline__ uint32_t rotl32(uint32_t v, int r) {
  return (v << r) | (v >> (32 - r));
}

__device__ __forceinline__ void tf2x32(uint32_t k0, uint32_t k1,
                                       uint32_t c0, uint32_t c1,
                                       uint32_t& o0, uint32_t& o1) {
  const uint32_t ks2 = k0 ^ k1 ^ 0x1BD11BDAu;
  uint32_t x0 = c0 + k0, x1 = c1 + k1;
#define TFR(r) { x0 += x1; x1 = rotl32(x1, (r)); x1 ^= x0; }
  TFR(13) TFR(15) TFR(26) TFR(6)   x0 += k1;  x1 += ks2 + 1u;
  TFR(17) TFR(29) TFR(16) TFR(24)  x0 += ks2; x1 += k0 + 2u;
  TFR(13) TFR(15) TFR(26) TFR(6)   x0 += k0;  x1 += k1 + 3u;
  TFR(17) TFR(29) TFR(16) TFR(24)  x0 += k1;  x1 += ks2 + 4u;
  TFR(13) TFR(15) TFR(26) TFR(6)   x0 += ks2; x1 += k0 + 5u;
#undef TFR
  o0 = x0; o1 = x1;
}

// jax_threefry_partitionable=True (default since JAX 0.4.36):
//   split(key)[i]         = threefry(key, hi=0, lo=i) -> child key (x0, x1)
//   random_bits32(key)[j] = x0 ^ x1 of threefry(key, 0, j)
__device__ __forceinline__ void tf_child(uint32_t k0, uint32_t k1, uint32_t i,
                                         uint32_t& c0, uint32_t& c1) {
  tf2x32(k0, k1, 0u, i, c0, c1);
}

__device__ __forceinline__ uint32_t tf_bits32(uint32_t k0, uint32_t k1, uint32_t j) {
  uint32_t o0, o1;
  tf2x32(k0, k1, 0u, j, o0, o1);
  return o0 ^ o1;
}

__device__ __forceinline__ void band_params(uint32_t k0, uint32_t k1, int j,
                                            int size, uint32_t span,
                                            int& lo_out, int& hi_out) {
  uint32_t kw0, kw1, ku0, ku1, k10, k11, k20, k21;
  tf_child(k0, k1, 0u, kw0, kw1);            // k_w
  tf_child(k0, k1, 1u, ku0, ku1);            // k_u
  tf_child(kw0, kw1, 0u, k10, k11);          // randint draw keys
  tf_child(kw0, kw1, 1u, k20, k21);
  const uint32_t hib   = tf_bits32(k10, k11, (uint32_t)j);
  const uint32_t lob   = tf_bits32(k20, k21, (uint32_t)j);
  const uint32_t ubits = tf_bits32(ku0, ku1, (uint32_t)j);

  // randint: ((hi % span) * (2^32 mod span) + lo % span) % span
  uint32_t mult = 65536u % span;
  mult = (mult * mult) % span;
  const uint32_t w = ((hib % span) * mult + (lob % span)) % span;

  // uniform [0,1): bitcast((bits >> 9) | 0x3f800000) - 1.0
  const float u = __uint_as_float((ubits >> 9) | 0x3f800000u) - 1.0f;
  const int w0 = (int)floorf(u * (float)(size - (int)w));
  lo_out = w0;
  hi_out = w0 + (int)w;
}

// Kernel B: thread j = batch row j -> {f_lo, f_hi, t_lo, t_hi} in d_ws.
__global__ __launch_bounds__(128)
void specaug_params_kernel(int* __restrict__ prm) {
  const int j = threadIdx.x;
  uint32_t kf0, kf1, kt0, kt1;
  tf_child(0u, 42u, 0u, kf0, kf1);   // kf, kt = split(key(42))
  tf_child(0u, 42u, 1u, kt0, kt1);

  int flo, fhi, tlo, thi;
  band_params(kf0, kf1, j, F_DIM, F_MAG, flo, fhi);
  band_params(kt0, kt1, j, T_DIM, T_MAG, tlo, thi);

  prm[4 * j + 0] = flo;
  prm[4 * j + 1] = fhi;
  prm[4 * j + 2] = tlo;
  prm[4 * j + 3] = thi;
}

extern "C" void kernel_launch(void* const* d_in, const int* in_sizes, int n_in,
                              void* d_out, int out_size, void* d_ws, size_t ws_size,
                              hipStream_t stream) {
  (void)in_sizes; (void)n_in; (void)out_size; (void)ws_size;
  const float* x = (const float*)d_in[0];
  float* out = (float*)d_out;
  int* prm = (int*)d_ws;   // 2 KB scratch

  specaug_params_kernel<<<1, 128, 0, stream>>>(prm);
  specaug_apply_kernel<<<APPLY_BLOCKS, APPLY_THREADS, 0, stream>>>(
      x, (f4*)out, prm);
}